// CNF_16501264351447
// MI455X (gfx1250) — compile-verified
//
#include <hip/hip_runtime.h>
#include <math.h>

typedef __attribute__((ext_vector_type(2))) float v2f;
typedef __attribute__((ext_vector_type(8))) float v8f;
typedef unsigned int u32x4 __attribute__((ext_vector_type(4)));
typedef int i32x4 __attribute__((ext_vector_type(4)));
typedef int i32x8 __attribute__((ext_vector_type(8)));

#define TSCALE 0.01f

#if defined(__gfx1250__) && __has_builtin(__builtin_amdgcn_tensor_load_to_lds)
#define USE_TDM 1
#else
#define USE_TDM 0
#endif

// ---------------- LDS layout (in floats) ----------------
// Workgroup tile: 32 samples -> 96 rows (32 primal, 32 tangent-u, 32 tangent-v)
// act row stride 260 => A-fragment ds_load_b64 hits 64 distinct banks.
#define NROWS      96
#define ACT_STRIDE 260
#define OFF_ACT    0
#define N_ACT      (NROWS * ACT_STRIDE)          // 24960
#define N_BSTB     4096                          // one 256x16 W2 column block
#define OFF_BST    (OFF_ACT + N_ACT)             // double-buffered: 2 * 4096
#define OFF_W1     (OFF_BST + 2 * N_BSTB)        // 3 x 256 row-major
#define OFF_B1     (OFF_W1 + 768)
#define OFF_B2     (OFF_B1 + 256)
#define OFF_W3     (OFF_B2 + 256)                // 256 x 2 row-major
#define OFF_B3     (OFF_W3 + 512)
#define OFF_XS     (OFF_B3 + 2)                  // stage input  [32][2]
#define OFF_XC     (OFF_XS + 64)                 // current x    [32][2]
#define OFF_AX     (OFF_XC + 64)                 // RK accum x   [32][2]
#define OFF_AL     (OFF_AX + 64)                 // RK accum ld  [32]
#define OFF_LD     (OFF_AL + 32)                 // logdet       [32]
#define OFF_FX     (OFF_LD + 32)                 // f(x,t)       [32][2]
#define OFF_TRU    (OFF_FX + 64)                 // J[0,0] part  [32]
#define OFF_TRV    (OFF_TRU + 32)                // J[1,1] part  [32]
#define SMEM_FLOATS (OFF_TRV + 32)
#define SMEM_BYTES  (SMEM_FLOATS * 4)            // ~138 KB -> 2 WGs per 320KB WGP

#if USE_TDM
// Issue a TDM DMA of W2 columns [16nt, 16nt+16) (256 rows x 16 f32 tile,
// row-major) into LDS at dst. Descriptor per CDNA5 ISA D# spec.
__device__ __forceinline__ void tdm_load_w2_block(const float* __restrict__ W2g,
                                                  int nt, float* dst) {
    unsigned lds_addr = (unsigned)(uintptr_t)dst;  // low 32 bits of flat = LDS addr
    unsigned long long ga = (unsigned long long)(uintptr_t)(W2g + nt * 16);
    u32x4 g0;
    g0.x = 1u;                                   // count=1, user descriptor
    g0.y = lds_addr;                             // lds_addr [63:32]
    g0.z = (unsigned)(ga & 0xffffffffu);         // global_addr low
    g0.w = (unsigned)((ga >> 32) & 0x01ffffffu)  // global_addr[56:32]
           | (2u << 30);                         // type = 2 ("image")
    i32x8 g1;
    g1[0] = (2 << 16);                           // data_size = 4B; no mask/pad/iter
    g1[1] = (int)(256u << 16);                   // tensor_dim0 = 256 (low16)
    g1[2] = (int)(256u << 16);                   // dim0 hi=0 | tensor_dim1=256 (low16)
    g1[3] = (int)(16u << 16);                    // dim1 hi=0 | tile_dim0 = 16
    g1[4] = 256;                                 // tile_dim1 = 256, tile_dim2 = 0
    g1[5] = 256;                                 // tensor_dim0_stride = 256 (low32)
    g1[6] = 0;                                   // stride0 hi | stride1 low
    g1[7] = 0;
    i32x4 z4 = {0, 0, 0, 0};
#if __clang_major__ >= 23
    i32x8 z8 = {0, 0, 0, 0, 0, 0, 0, 0};
    __builtin_amdgcn_tensor_load_to_lds(g0, g1, z4, z4, z8, 0);
#else
    __builtin_amdgcn_tensor_load_to_lds(g0, g1, z4, z4, 0);
#endif
}

__device__ __forceinline__ void wait_tensor0() {
#if __has_builtin(__builtin_amdgcn_s_wait_tensorcnt)
    __builtin_amdgcn_s_wait_tensorcnt(0);
#else
    asm volatile("s_wait_tensorcnt 0x0" ::: "memory");
#endif
}
#endif  // USE_TDM

// One aug() evaluation: fills fx / trU / trV from xs at time t.
__device__ __forceinline__ void aug_eval(float* __restrict__ sm,
                                         const float* __restrict__ W2g,
                                         float t, int tid, int lane, int lhalf,
                                         int col, int wid, int rowbase)
{
    float* act = sm + OFF_ACT;
    float* bst = sm + OFF_BST;
    const float* w1  = sm + OFF_W1;
    const float* b1s = sm + OFF_B1;
    const float* b2s = sm + OFF_B2;
    const float* w3  = sm + OFF_W3;
    const float* b3s = sm + OFF_B3;
    const float* xs  = sm + OFF_XS;
    float* fx  = sm + OFF_FX;
    float* trU = sm + OFF_TRU;
    float* trV = sm + OFF_TRV;

    // ---- Layer 1 (K=3) + tanh, tangent seeds computed analytically ----
    // primal:    h1 = tanh(x0*W1[0] + x1*W1[1] + t*W1[2] + b1)
    // tangent-u: u1 = W1[0] * (1 - h1^2)    tangent-v: v1 = W1[1] * (1 - h1^2)
    for (int i = tid; i < 32 * 256; i += 192) {
        int s = i >> 8;
        int n = i & 255;
        float wa = w1[n], wb = w1[256 + n], wc = w1[512 + n];
        float pre = xs[2*s] * wa + xs[2*s+1] * wb + t * wc + b1s[n];
        float hv  = tanhf(pre);
        float g   = 1.0f - hv * hv;
        act[s * ACT_STRIDE + n]        = hv;
        act[(32 + s) * ACT_STRIDE + n] = wa * g;
        act[(64 + s) * ACT_STRIDE + n] = wb * g;
    }

    // ---- Layer 2: [96 x 256] @ W2[256 x 256] via V_WMMA_F32_16X16X4_F32 ----
    v8f acc[16];
    const v8f vz = {0.f,0.f,0.f,0.f,0.f,0.f,0.f,0.f};
#pragma unroll
    for (int nt = 0; nt < 16; ++nt) acc[nt] = vz;

#if USE_TDM
    if (wid == 0) tdm_load_w2_block(W2g, 0, bst);   // prologue: DMA block 0
#endif

#pragma unroll
    for (int nt = 0; nt < 16; ++nt) {
        float* bstc = bst + (nt & 1) * N_BSTB;
#if USE_TDM
        if (wid == 0) wait_tensor0();               // block nt landed in LDS
        __syncthreads();                            // publish; prior compute done
        if (wid == 0 && nt < 15)                    // DMA next block during compute
            tdm_load_w2_block(W2g, nt + 1, bst + ((nt + 1) & 1) * N_BSTB);
#else
        __syncthreads();
        for (int i = tid; i < 4096; i += 192)       // cooperative staging fallback
            bstc[i] = W2g[(i >> 4) * 256 + nt * 16 + (i & 15)];
        __syncthreads();
#endif
        // A frag: lanes 0-15 = rows M, halves give K pairs {4k+2h, 4k+2h+1}
        const float* aptr = act + (rowbase + col) * ACT_STRIDE + 2 * lhalf;
        // B frag (row-major block): b.x = W2[4k+2h][col], b.y = W2[4k+2h+1][col]
        const float* bptr = bstc + lhalf * 32 + col;
#pragma unroll 4
        for (int k = 0; k < 64; ++k) {
            v2f a = *(const v2f*)(aptr + 4 * k);
            v2f b;
            b.x = bptr[64 * k];
            b.y = bptr[64 * k + 16];
            acc[nt] = __builtin_amdgcn_wmma_f32_16x16x4_f32(
                false, a, false, b, (short)0, acc[nt], false, false);
        }
    }
    __syncthreads();

    // ---- Elementwise: primal waves tanh(.. + b2), then tangent waves scale ----
    if (wid < 2) {  // rows 0..31: primal
#pragma unroll
        for (int nt = 0; nt < 16; ++nt) {
            float bb = b2s[nt * 16 + col];
#pragma unroll
            for (int v = 0; v < 8; ++v) {
                int m = v + 8 * lhalf;               // C/D layout: M = v + 8*half
                float hv = tanhf(acc[nt][v] + bb);
                act[(rowbase + m) * ACT_STRIDE + nt * 16 + col] = hv;
            }
        }
    }
    __syncthreads();
    if (wid >= 2) {  // tangent rows: t2 = mid * (1 - h2^2) of matching primal row
        int pbase = rowbase - ((wid >= 4) ? 64 : 32);
#pragma unroll
        for (int nt = 0; nt < 16; ++nt) {
#pragma unroll
            for (int v = 0; v < 8; ++v) {
                int m = v + 8 * lhalf;
                float hp = act[(pbase + m) * ACT_STRIDE + nt * 16 + col];
                act[(rowbase + m) * ACT_STRIDE + nt * 16 + col] =
                    acc[nt][v] * (1.0f - hp * hp);
            }
        }
    }
    __syncthreads();

    // ---- Layer 3 (N=2): 256-wide dots via wave reductions ----
    for (int task = wid; task < 128; task += 6) {
        int s    = task & 31;
        int comp = task >> 5;
        int row  = (comp < 2) ? s : ((comp == 2) ? (32 + s) : (64 + s));
        int wc   = (comp == 1 || comp == 3) ? 1 : 0;
        float p = 0.f;
        for (int i = lane; i < 256; i += 32)
            p += act[row * ACT_STRIDE + i] * w3[2 * i + wc];
#pragma unroll
        for (int off = 16; off > 0; off >>= 1)
            p += __shfl_xor(p, off, 32);
        if (lane == 0) {
            if (comp < 2)        fx[2 * s + comp] = p + b3s[comp];
            else if (comp == 2)  trU[s] = p;
            else                 trV[s] = p;
        }
    }
    __syncthreads();
}

__global__ __launch_bounds__(192)
void cnf_rk4_wmma(const float* __restrict__ xg,
                  const float* __restrict__ W1g, const float* __restrict__ b1g,
                  const float* __restrict__ W2g, const float* __restrict__ b2g,
                  const float* __restrict__ W3g, const float* __restrict__ b3g,
                  const int*   __restrict__ nstepsp,
                  float* __restrict__ outg, int Btot)
{
    extern __shared__ float sm[];
    const int tid   = threadIdx.x;
    const int lane  = tid & 31;
    const int lhalf = (lane >> 4) & 1;
    const int col   = lane & 15;
    const int wid   = tid >> 5;
    const int rowbase = wid * 16;

    // Preload small weights into LDS (W2 streams from L2 per tile via TDM).
    for (int i = tid; i < 768; i += 192) sm[OFF_W1 + i] = W1g[i];
    for (int i = tid; i < 256; i += 192) { sm[OFF_B1 + i] = b1g[i]; sm[OFF_B2 + i] = b2g[i]; }
    for (int i = tid; i < 512; i += 192) sm[OFF_W3 + i] = W3g[i];
    if (tid < 2) sm[OFF_B3 + tid] = b3g[tid];
    if (tid < 32) {
        int g = blockIdx.x * 32 + tid;
        float x0 = 0.f, x1 = 0.f;
        if (g < Btot) { x0 = xg[2 * g]; x1 = xg[2 * g + 1]; }
        sm[OFF_XC + 2 * tid] = x0; sm[OFF_XC + 2 * tid + 1] = x1;
        sm[OFF_XS + 2 * tid] = x0; sm[OFF_XS + 2 * tid + 1] = x1;
        sm[OFF_LD + tid] = 0.f;
    }
    __syncthreads();

    const int nsteps = *nstepsp;
    const float hstep = 1.0f / (float)(nsteps - 1);

    for (int step = 0; step < nsteps - 1; ++step) {
        float t0 = (float)step * hstep;
#pragma unroll 1
        for (int st = 0; st < 4; ++st) {
            float t = t0 + ((st == 0) ? 0.f : (st == 3) ? hstep : 0.5f * hstep);
            aug_eval(sm, W2g, t, tid, lane, lhalf, col, wid, rowbase);

            if (tid < 32) {  // RK4 bookkeeping, one thread per sample
                float fl = -TSCALE * (sm[OFF_TRU + tid] + sm[OFF_TRV + tid]);
                float f0 = sm[OFF_FX + 2 * tid], f1 = sm[OFF_FX + 2 * tid + 1];
                float w  = (st == 0 || st == 3) ? 1.f : 2.f;
                if (st == 0) {
                    sm[OFF_AX + 2 * tid] = f0; sm[OFF_AX + 2 * tid + 1] = f1;
                    sm[OFF_AL + tid] = fl;
                } else {
                    sm[OFF_AX + 2 * tid] += w * f0; sm[OFF_AX + 2 * tid + 1] += w * f1;
                    sm[OFF_AL + tid] += w * fl;
                }
                if (st < 3) {
                    float c = (st == 2) ? hstep : 0.5f * hstep;
                    sm[OFF_XS + 2 * tid]     = sm[OFF_XC + 2 * tid]     + c * f0;
                    sm[OFF_XS + 2 * tid + 1] = sm[OFF_XC + 2 * tid + 1] + c * f1;
                } else {
                    float s6 = hstep * (1.0f / 6.0f);
                    float nx0 = sm[OFF_XC + 2 * tid]     + s6 * sm[OFF_AX + 2 * tid];
                    float nx1 = sm[OFF_XC + 2 * tid + 1] + s6 * sm[OFF_AX + 2 * tid + 1];
                    sm[OFF_XC + 2 * tid] = nx0; sm[OFF_XC + 2 * tid + 1] = nx1;
                    sm[OFF_XS + 2 * tid] = nx0; sm[OFF_XS + 2 * tid + 1] = nx1;
                    sm[OFF_LD + tid] += s6 * sm[OFF_AL + tid];
                }
            }
            __syncthreads();
        }
    }

    if (tid < 32) {
        int g = blockIdx.x * 32 + tid;
        if (g < Btot) {
            outg[2 * g]     = sm[OFF_XC + 2 * tid];
            outg[2 * g + 1] = sm[OFF_XC + 2 * tid + 1];
            outg[2 * Btot + g] = sm[OFF_LD + tid];
        }
    }
}

extern "C" void kernel_launch(void* const* d_in, const int* in_sizes, int n_in,
                              void* d_out, int out_size, void* d_ws, size_t ws_size,
                              hipStream_t stream) {
    (void)n_in; (void)out_size; (void)d_ws; (void)ws_size;
    const float* x   = (const float*)d_in[0];
    const float* W1  = (const float*)d_in[1];
    const float* b1  = (const float*)d_in[2];
    const float* W2  = (const float*)d_in[3];
    const float* b2  = (const float*)d_in[4];
    const float* W3  = (const float*)d_in[5];
    const float* b3  = (const float*)d_in[6];
    const int* nst   = (const int*)d_in[7];
    int Btot = in_sizes[0] / 2;
    int blocks = (Btot + 31) / 32;

    // Deterministic, idempotent host-side attribute (not a stream op; capture-safe).
    hipFuncSetAttribute(reinterpret_cast<const void*>(cnf_rk4_wmma),
                        hipFuncAttributeMaxDynamicSharedMemorySize, SMEM_BYTES);

    cnf_rk4_wmma<<<blocks, 192, SMEM_BYTES, stream>>>(
        x, W1, b1, W2, b2, W3, b3, nst, (float*)d_out, Btot);
}